// FlexibleNet_1924145349062
// MI455X (gfx1250) — compile-verified
//
#include <hip/hip_runtime.h>
#include <math.h>

#define N_NODES 102400
#define FEAT    50
#define DIM     128
#define NHEAD   4
#define NEDGE   1638400
#define NPG     100
#define NGRAPH  1024
#define NSEED1  75
#define INV_SCALE 0.08838834764831843f   /* 1/sqrt(128) */

typedef float v2f __attribute__((ext_vector_type(2)));
typedef float v8f __attribute__((ext_vector_type(8)));

// -------------------------------------------------------------------------
// FAST WMMA f32 GEMM: K == 128, M % 128 == 0, no guards anywhere.
// out[M,128] = epi( A[M,128]@W[128,128] (+ A2@W2) + bias )
// epi: flags&1 -> relu ; flags&2 -> add resid[M,128] AFTER relu (MAB epilogue)
// Wave = 16 rows x 128 cols (8 accum tiles). Block = 256 thr = 8 waves.
//
// W staged in LDS with a pair-interleaved parity-swizzled layout:
//   element (row, col) -> word (row/2)*256 + ((2col) ^ (((row/2)&1)<<5)) + (row&1)
// so each B-fragment (W[br][bc], W[br+1][bc]) is one contiguous ds_load_b64,
// and the two lane-halves (pairIdx parity == lh) hit disjoint bank halves.
// -------------------------------------------------------------------------
__global__ __launch_bounds__(256) void gemm128_f(
    const float* __restrict__ A,  const float* __restrict__ W,
    const float* __restrict__ A2, const float* __restrict__ W2,
    const float* __restrict__ bias, const float* __restrict__ resid,
    float* __restrict__ out, int M, int flags)
{
  __shared__ float sW[128 * 128];          // 64 KB
  const int t    = threadIdx.x;
  const int wave = t >> 5;
  const int lane = t & 31;
  const int lh   = lane >> 4;   // K sub-pair select (A/B), M+8 select (C/D)
  const int ll   = lane & 15;   // row (A) / col (B/C/D)
  const int row0 = blockIdx.x * 128 + wave * 16;

  // swizzled staging (coalesced global reads, conflict-free LDS writes)
#pragma unroll 4
  for (int i = t; i < 128 * 128; i += 256) {
    const int row  = i >> 7;
    const int col  = i & 127;
    const int pair = row >> 1;
    const int w    = pair * 256 + (((col << 1) ^ ((pair & 1) << 5))) + (row & 1);
    sW[w] = W[i];
  }
  __syncthreads();

  // 8 per-lane fragment base pointers (constant across the K loop)
  const float* bp[8];
#pragma unroll
  for (int tn = 0; tn < 8; ++tn)
    bp[tn] = sW + lh * 256 + ((tn ^ lh) << 5) + (ll << 1);

  v8f acc[8] = {};

  const float* Arow = A + (long)(row0 + ll) * 128 + lh * 2;
  for (int k = 0; k < 128; k += 4) {
    const v2f a = *(const v2f*)(Arow + k);          // k, k+1 contiguous
    const int ko = (k >> 1) * 256;                  // immediate DS offset
#pragma unroll
    for (int tn = 0; tn < 8; ++tn) {
      const v2f b = *(const v2f*)(bp[tn] + ko);     // single ds_load_b64
      acc[tn] = __builtin_amdgcn_wmma_f32_16x16x4_f32(
          false, a, false, b, (short)0, acc[tn], false, false);
    }
  }
  if (A2) {
    __syncthreads();
#pragma unroll 4
    for (int i = t; i < 128 * 128; i += 256) {
      const int row  = i >> 7;
      const int col  = i & 127;
      const int pair = row >> 1;
      const int w    = pair * 256 + (((col << 1) ^ ((pair & 1) << 5))) + (row & 1);
      sW[w] = W2[i];
    }
    __syncthreads();
    const float* A2row = A2 + (long)(row0 + ll) * 128 + lh * 2;
    for (int k = 0; k < 128; k += 4) {
      const v2f a = *(const v2f*)(A2row + k);
      const int ko = (k >> 1) * 256;
#pragma unroll
      for (int tn = 0; tn < 8; ++tn) {
        const v2f b = *(const v2f*)(bp[tn] + ko);
        acc[tn] = __builtin_amdgcn_wmma_f32_16x16x4_f32(
            false, a, false, b, (short)0, acc[tn], false, false);
      }
    }
  }

#pragma unroll
  for (int tn = 0; tn < 8; ++tn) {
    const int col = tn * 16 + ll;
    const float bv = bias ? bias[col] : 0.0f;
#pragma unroll
    for (int r = 0; r < 8; ++r) {
      const long row = row0 + r + lh * 8;   // C layout: VGPR r -> M=r / M=r+8
      float v = acc[tn][r] + bv;
      if (flags & 1) v = fmaxf(v, 0.0f);
      if (flags & 2) v += resid[row * 128 + col];
      out[row * 128 + col] = v;
    }
  }
}

// -------------------------------------------------------------------------
// Guarded WMMA GEMM for odd shapes (K=50 input layer; M=75 / M=1 seeds).
// Branch-free guards: clamp the address, always load, select 0 (v_cndmask).
// -------------------------------------------------------------------------
__global__ __launch_bounds__(256) void gemm128_g(
    const float* __restrict__ A, const float* __restrict__ W,
    const float* __restrict__ bias, float* __restrict__ out,
    int M, int K, int flags)
{
  const int t    = threadIdx.x;
  const int wave = t >> 5;
  const int lane = t & 31;
  const int lh   = lane >> 4;
  const int ll   = lane & 15;
  const int row0 = blockIdx.x * 128 + wave * 16;

  v8f acc[8] = {};

  const int ar   = row0 + ll;
  const int arc  = (ar < M) ? ar : (M - 1);        // clamped row
  const bool rok = (ar < M);

  for (int k = 0; k < K; k += 4) {
    const int ac  = k + lh * 2;
    const int ac0 = (ac     < K) ? ac     : (K - 1);
    const int ac1 = (ac + 1 < K) ? ac + 1 : (K - 1);
    v2f a;
    a.x = A[(long)arc * K + ac0];
    a.y = A[(long)arc * K + ac1];
    a.x = (rok && ac     < K) ? a.x : 0.0f;
    a.y = (rok && ac + 1 < K) ? a.y : 0.0f;
#pragma unroll
    for (int tn = 0; tn < 8; ++tn) {
      const int bc = tn * 16 + ll;
      v2f b;
      b.x = W[ac0 * 128 + bc];
      b.y = W[ac1 * 128 + bc];
      b.x = (ac     < K) ? b.x : 0.0f;
      b.y = (ac + 1 < K) ? b.y : 0.0f;
      acc[tn] = __builtin_amdgcn_wmma_f32_16x16x4_f32(
          false, a, false, b, (short)0, acc[tn], false, false);
    }
  }

#pragma unroll
  for (int tn = 0; tn < 8; ++tn) {
    const int col = tn * 16 + ll;
    const float bv = bias ? bias[col] : 0.0f;
#pragma unroll
    for (int r = 0; r < 8; ++r) {
      const int row = row0 + r + lh * 8;
      if (row < M) {
        float v = acc[tn][r] + bv;
        if (flags & 1) v = fmaxf(v, 0.0f);
        out[(long)row * 128 + col] = v;
      }
    }
  }
}

// ------------------------- edge / segment kernels -------------------------
__global__ __launch_bounds__(256) void count_k(const int* __restrict__ dst,
                                               float* __restrict__ cnt, int E)
{
  int i = blockIdx.x * blockDim.x + threadIdx.x;
  if (i < E) atomicAdd(&cnt[dst[i]], 1.0f);
}

__global__ __launch_bounds__(256) void dis_k(const float* __restrict__ cnt,
                                             float* __restrict__ dis, int n)
{
  int i = blockIdx.x * blockDim.x + threadIdx.x;
  if (i < n) dis[i] = rsqrtf(1.0f + cnt[i]);     // deg = 1 + indeg (self loop)
}

// one wave per edge, lane handles 4 consecutive columns (L2-resident atomics)
__global__ __launch_bounds__(256) void sage_scatter_k(
    const float* __restrict__ h, const int* __restrict__ src,
    const int* __restrict__ dst, float* __restrict__ s, int E)
{
  long t = (long)blockIdx.x * blockDim.x + threadIdx.x;
  long e = t >> 5;
  int  l = (int)(t & 31);
  if (e >= E) return;
  const int si = src[e], di = dst[e];
  const float4 v = *(const float4*)(h + (long)si * 128 + l * 4);
  float* o = s + (long)di * 128 + l * 4;
  atomicAdd(o + 0, v.x); atomicAdd(o + 1, v.y);
  atomicAdd(o + 2, v.z); atomicAdd(o + 3, v.w);
}

__global__ __launch_bounds__(256) void gcn_scatter_k(
    const float* __restrict__ hw, const int* __restrict__ src,
    const int* __restrict__ dst, const float* __restrict__ dis,
    float* __restrict__ out, int E)
{
  long t = (long)blockIdx.x * blockDim.x + threadIdx.x;
  long e = t >> 5;
  int  l = (int)(t & 31);
  if (e >= E) return;
  const int si = src[e], di = dst[e];
  const float sc = dis[si] * dis[di];
  const float4 v = *(const float4*)(hw + (long)si * 128 + l * 4);
  float* o = out + (long)di * 128 + l * 4;
  atomicAdd(o + 0, v.x * sc); atomicAdd(o + 1, v.y * sc);
  atomicAdd(o + 2, v.z * sc); atomicAdd(o + 3, v.w * sc);
}

__global__ __launch_bounds__(256) void rowdiv_k(float* __restrict__ s,
                                                const float* __restrict__ cnt,
                                                long n128)
{
  long i = (long)blockIdx.x * blockDim.x + threadIdx.x;
  if (i < n128) {
    long r = i >> 7;
    s[i] *= 1.0f / fmaxf(cnt[r], 1.0f);
  }
}

__global__ __launch_bounds__(256) void gcn_self_k(
    const float* __restrict__ hw, const float* __restrict__ dis,
    const float* __restrict__ bias, float* __restrict__ out, long n128)
{
  long i = (long)blockIdx.x * blockDim.x + threadIdx.x;
  if (i < n128) {
    long r = i >> 7;
    int  c = (int)(i & 127);
    const float d = dis[r];
    out[i] += hw[i] * d * d + bias[c];
  }
}

// ------------------------- MAB attention core -----------------------------
// block = 128 thr (4 waves, one per head) handles one (graph b, query q).
// out = Qp + softmax(Qh Kh^T / sqrt(D)) Vh   (residual is the projected Q)
__global__ __launch_bounds__(128) void mab_attn(
    const float* __restrict__ Q, long qbs,
    const float* __restrict__ K, const float* __restrict__ V,
    float* __restrict__ out, int nq, int nk)
{
  __shared__ float qs[128];
  __shared__ float sc[NHEAD][128];
  const int b  = blockIdx.x / nq;
  const int q  = blockIdx.x % nq;
  const int t  = threadIdx.x;
  const int hd = t >> 5;
  const int l  = t & 31;
  const float* Qrow = Q + (long)b * qbs + (long)q * 128;
  qs[t] = Qrow[t];
  __syncthreads();
  const float* Kbase = K + (long)b * nk * 128;
  for (int k = l; k < nk; k += 32) {
    const float* Krow = Kbase + (long)k * 128 + hd * 32;
    float s = 0.0f;
#pragma unroll
    for (int d = 0; d < 32; ++d) s += qs[hd * 32 + d] * Krow[d];
    sc[hd][k] = s * INV_SCALE;
  }
  __syncthreads();
  float m = -3.0e38f;
  for (int k = l; k < nk; k += 32) m = fmaxf(m, sc[hd][k]);
  for (int off = 16; off > 0; off >>= 1) m = fmaxf(m, __shfl_xor(m, off));
  float sum = 0.0f;
  for (int k = l; k < nk; k += 32) {
    const float e = __expf(sc[hd][k] - m);
    sc[hd][k] = e;
    sum += e;
  }
  for (int off = 16; off > 0; off >>= 1) sum += __shfl_xor(sum, off);
  const float inv = 1.0f / sum;
  __syncthreads();
  const float* Vbase = V + (long)b * nk * 128 + hd * 32 + l;
  float acc = 0.0f;
  for (int k = 0; k < nk; ++k) acc += sc[hd][k] * Vbase[(long)k * 128];
  out[((long)b * nq + q) * 128 + t] = qs[t] + acc * inv;
}

// ------------------------- final lin2 dot ---------------------------------
__global__ __launch_bounds__(256) void lin2_k(
    const float* __restrict__ o, const float* __restrict__ w,
    const float* __restrict__ bias, float* __restrict__ out, int Bn)
{
  int t  = blockIdx.x * blockDim.x + threadIdx.x;
  int bi = t >> 5;
  int l  = t & 31;
  if (bi >= Bn) return;
  float s = 0.0f;
  for (int c = l; c < 128; c += 32) s += o[(long)bi * 128 + c] * w[c];
  for (int off = 16; off > 0; off >>= 1) s += __shfl_xor(s, off);
  if (l == 0) out[bi] = s + bias[0];
}

// ==========================================================================
extern "C" void kernel_launch(void* const* d_in, const int* in_sizes, int n_in,
                              void* d_out, int out_size, void* d_ws, size_t ws_size,
                              hipStream_t stream)
{
  (void)in_sizes; (void)n_in; (void)out_size; (void)ws_size;
  // ---- inputs (setup_inputs dict order) ----
  const float* x       = (const float*)d_in[0];
  /* d_in[1] = batch (unused: graphs are uniform blocks of NPG nodes) */
  const float* w_in    = (const float*)d_in[2];
  const float* b_in    = (const float*)d_in[3];
  const float* sage_wl = (const float*)d_in[4];
  const float* sage_bl = (const float*)d_in[5];
  const float* sage_wr = (const float*)d_in[6];
  const float* lin1_w  = (const float*)d_in[7];
  const float* lin1_b  = (const float*)d_in[8];
  const float* p1_seed = (const float*)d_in[9];
  const float* p3_seed = (const float*)d_in[10];
  const float* lin2_w  = (const float*)d_in[11];
  const float* lin2_b  = (const float*)d_in[12];
  const float* p1_wq = (const float*)d_in[13], *p1_bq = (const float*)d_in[14];
  const float* p1_wk = (const float*)d_in[15], *p1_bk = (const float*)d_in[16];
  const float* p1_wv = (const float*)d_in[17], *p1_bv = (const float*)d_in[18];
  const float* p1_wo = (const float*)d_in[19], *p1_bo = (const float*)d_in[20];
  const float* s2_wq = (const float*)d_in[21], *s2_bq = (const float*)d_in[22];
  const float* s2_wk = (const float*)d_in[23], *s2_bk = (const float*)d_in[24];
  const float* s2_wv = (const float*)d_in[25], *s2_bv = (const float*)d_in[26];
  const float* s2_wo = (const float*)d_in[27], *s2_bo = (const float*)d_in[28];
  const float* p3_wq = (const float*)d_in[29], *p3_bq = (const float*)d_in[30];
  const float* p3_wk = (const float*)d_in[31], *p3_bk = (const float*)d_in[32];
  const float* p3_wv = (const float*)d_in[33], *p3_bv = (const float*)d_in[34];
  const float* p3_wo = (const float*)d_in[35], *p3_bo = (const float*)d_in[36];
  const int*   eidx  = (const int*)d_in[37];
  const int*   src   = eidx;
  const int*   dst   = eidx + NEDGE;

  // ---- workspace layout ----
  const size_t NBYTES = (size_t)N_NODES * 128 * sizeof(float);  // 52.4 MB
  char* base = (char*)d_ws;
  float* b0  = (float*)(base);
  float* b1  = (float*)(base + 1 * NBYTES);
  float* b2  = (float*)(base + 2 * NBYTES);
  float* b3  = (float*)(base + 3 * NBYTES);
  float* cnt = (float*)(base + 4 * NBYTES);
  float* dis = cnt + N_NODES;
  float* qp1 = dis + N_NODES;                 // 75 x 128
  float* qp3 = qp1 + NSEED1 * 128;            // 1 x 128

  const long N128   = (long)N_NODES * 128;
  const int  GN     = N_NODES / 128;          // 800 blocks
  const int  GROWS  = NGRAPH * NSEED1;        // 76800 rows
  const int  GROWSB = GROWS / 128;            // 600 blocks
  const int  EB     = (NEDGE * 32) / 256;     // 204800 blocks
  const int  NWB    = (int)((N128 + 255) / 256);

  // ---- in-degree counts + GCN normalization ----
  hipMemsetAsync(cnt, 0, N_NODES * sizeof(float), stream);
  count_k<<<(NEDGE + 255) / 256, 256, 0, stream>>>(dst, cnt, NEDGE);
  dis_k<<<(N_NODES + 255) / 256, 256, 0, stream>>>(cnt, dis, N_NODES);

  // ---- h = relu(x @ w_in + b_in)  (K=50 -> guarded kernel) ----
  gemm128_g<<<GN, 256, 0, stream>>>(x, w_in, b_in, b0, N_NODES, FEAT, 1);
  float* h  = b0;
  float* hn = b2;

  // ---- 3 x SAGEConv: relu(mean @ wl + bl + h @ wr) ----
  for (int l = 0; l < 3; ++l) {
    hipMemsetAsync(b1, 0, NBYTES, stream);
    sage_scatter_k<<<EB, 256, 0, stream>>>(h, src, dst, b1, NEDGE);
    rowdiv_k<<<NWB, 256, 0, stream>>>(b1, cnt, N128);
    gemm128_f<<<GN, 256, 0, stream>>>(b1, sage_wl + (size_t)l * 128 * 128,
                                      h,  sage_wr + (size_t)l * 128 * 128,
                                      sage_bl + (size_t)l * 128, nullptr,
                                      hn, N_NODES, 1);
    float* tmp = h; h = hn; hn = tmp;
  }
  // h = b2 after 3 swaps; hn = b0

  // ---- lin1 ----
  gemm128_f<<<GN, 256, 0, stream>>>(h, lin1_w, nullptr, nullptr, lin1_b,
                                    nullptr, hn, N_NODES, 0);
  float* hL    = hn;   // b0
  float* freeb = h;    // b2

  // ---- GCNConv for K (p1) ----
  gemm128_f<<<GN, 256, 0, stream>>>(hL, p1_wk, nullptr, nullptr, nullptr,
                                    nullptr, b1, N_NODES, 0);          // hw
  hipMemsetAsync(freeb, 0, NBYTES, stream);
  gcn_scatter_k<<<EB, 256, 0, stream>>>(b1, src, dst, dis, freeb, NEDGE);
  gcn_self_k<<<NWB, 256, 0, stream>>>(b1, dis, p1_bk, freeb, N128);
  float* Kb = freeb;   // b2

  // ---- GCNConv for V (p1) ----
  gemm128_f<<<GN, 256, 0, stream>>>(hL, p1_wv, nullptr, nullptr, nullptr,
                                    nullptr, b1, N_NODES, 0);
  hipMemsetAsync(b3, 0, NBYTES, stream);
  gcn_scatter_k<<<EB, 256, 0, stream>>>(b1, src, dst, dis, b3, NEDGE);
  gcn_self_k<<<NWB, 256, 0, stream>>>(b1, dis, p1_bv, b3, N128);
  float* Vb = b3;

  // ---- Qp1 = p1_seed @ p1_wq + p1_bq  (shared across all graphs) ----
  gemm128_g<<<1, 256, 0, stream>>>(p1_seed, p1_wq, p1_bq, qp1, NSEED1, 128, 0);

  // ---- MAB 1 (GMPool_G): nq=75, nk=100 ----
  float* o1 = hL;  // b0 free after projections above
  mab_attn<<<NGRAPH * NSEED1, 128, 0, stream>>>(qp1, 0, Kb, Vb, o1, NSEED1, NPG);
  gemm128_f<<<GROWSB, 256, 0, stream>>>(o1, p1_wo, nullptr, nullptr, p1_bo, o1,
                                        b1, GROWS, 3);   // o = o1 + relu(.)
  float* o = b1;

  // ---- SAB (s2): Q,K,V projections of o ----
  float* Q2 = b2;  // Kb consumed
  float* K2 = b3;  // Vb consumed
  float* V2 = b0;  // o1 consumed
  gemm128_f<<<GROWSB, 256, 0, stream>>>(o, s2_wq, nullptr, nullptr, s2_bq, nullptr, Q2, GROWS, 0);
  gemm128_f<<<GROWSB, 256, 0, stream>>>(o, s2_wk, nullptr, nullptr, s2_bk, nullptr, K2, GROWS, 0);
  gemm128_f<<<GROWSB, 256, 0, stream>>>(o, s2_wv, nullptr, nullptr, s2_bv, nullptr, V2, GROWS, 0);
  float* o2 = b1;  // o consumed by projections
  mab_attn<<<NGRAPH * NSEED1, 128, 0, stream>>>(Q2, (long)NSEED1 * 128, K2, V2,
                                                o2, NSEED1, NSEED1);
  float* o2b = b0; // V2 consumed
  gemm128_f<<<GROWSB, 256, 0, stream>>>(o2, s2_wo, nullptr, nullptr, s2_bo, o2,
                                        o2b, GROWS, 3);

  // ---- MAB 3 (GMPool_I): 1 seed ----
  gemm128_g<<<1, 256, 0, stream>>>(p3_seed, p3_wq, p3_bq, qp3, 1, 128, 0);
  float* K3 = b2;
  float* V3 = b3;
  gemm128_f<<<GROWSB, 256, 0, stream>>>(o2b, p3_wk, nullptr, nullptr, p3_bk, nullptr, K3, GROWS, 0);
  gemm128_f<<<GROWSB, 256, 0, stream>>>(o2b, p3_wv, nullptr, nullptr, p3_bv, nullptr, V3, GROWS, 0);
  float* o3 = b1;
  mab_attn<<<NGRAPH, 128, 0, stream>>>(qp3, 0, K3, V3, o3, 1, NSEED1);
  float* o3b = b0;
  gemm128_f<<<NGRAPH / 128, 256, 0, stream>>>(o3, p3_wo, nullptr, nullptr,
                                              p3_bo, o3, o3b, NGRAPH, 3);

  // ---- lin2 -> [B] ----
  lin2_k<<<(NGRAPH * 32) / 256, 256, 0, stream>>>(o3b, lin2_w, lin2_b,
                                                  (float*)d_out, NGRAPH);
}